// AttentionPairBias_5531917877752
// MI455X (gfx1250) — compile-verified
//
#include <hip/hip_runtime.h>

#define DIM    256
#define NHEADS 8
#define HD     32
#define LSEQ   512
#define BATCH  2

typedef float v2f __attribute__((ext_vector_type(2)));
typedef float v8f __attribute__((ext_vector_type(8)));
typedef unsigned int u32x4 __attribute__((ext_vector_type(4)));
typedef int i32x8 __attribute__((ext_vector_type(8)));
typedef int i32x4 __attribute__((ext_vector_type(4)));

#if defined(__has_builtin)
#if __has_builtin(__builtin_amdgcn_tensor_load_to_lds) && \
    __has_builtin(__builtin_amdgcn_s_wait_tensorcnt)
#define USE_TDM 1
#endif
#endif
#ifndef USE_TDM
#define USE_TDM 0
#endif

__device__ __forceinline__ v8f wmma4(v2f a, v2f b, v8f c) {
  // V_WMMA_F32_16X16X4_F32 : D = A(16x4) * B(4x16) + C(16x16), all f32
  return __builtin_amdgcn_wmma_f32_16x16x4_f32(false, a, false, b, (short)0, c,
                                               false, false);
}

__device__ __forceinline__ unsigned lds_of(const void* p) {
  // LDS flat aperture: bits[31:0] are the LDS byte offset.
  return (unsigned)(unsigned long long)p;
}

#if USE_TDM
// 2D TDM load: tile (tile_d0 x tile_d1) of 4-byte elements, row stride
// `stride0` elements in memory, optional LDS padding (pad codes per ISA
// group1: interval code 4->32dw, 7->256dw; amount code = dwords-1).
__device__ __forceinline__ void tdm_load_2d(unsigned lds_addr, const void* gptr,
                                            unsigned tile_d0, unsigned tile_d1,
                                            unsigned tensor_d0,
                                            unsigned tensor_d1,
                                            unsigned stride0, int pad_en,
                                            unsigned pad_interval_code,
                                            unsigned pad_amount_code) {
  unsigned long long ga = (unsigned long long)gptr;
  u32x4 g0;
  g0.x = 1u;                                     // count=1 valid descriptor
  g0.y = lds_addr;                               // LDS byte address
  g0.z = (unsigned)(ga & 0xffffffffu);           // global_addr[31:0]
  g0.w = (unsigned)((ga >> 32) & 0x01ffffffu) |  // global_addr[56:32]
         (2u << 30);                             // type=2 ("image")
  unsigned dw0 = (2u << 16);                     // data_size: 4 bytes
  if (pad_en)
    dw0 |= (1u << 20) | (pad_interval_code << 22) | (pad_amount_code << 25);
  i32x8 g1;
  g1[0] = (int)dw0;
  g1[1] = (int)((tensor_d0 & 0xffffu) << 16);    // tensor_dim0[15:0] @ [31:16]
  g1[2] = (int)(((tensor_d0 >> 16) & 0xffffu) | ((tensor_d1 & 0xffffu) << 16));
  g1[3] = (int)(((tensor_d1 >> 16) & 0xffffu) | ((tile_d0 & 0xffffu) << 16));
  g1[4] = (int)(tile_d1 & 0xffffu);              // tile_dim1; tile_dim2=0
  g1[5] = (int)stride0;                          // tensor_dim0_stride[31:0]
  g1[6] = 0;                                     // stride0 hi / stride1 lo
  g1[7] = 0;
  i32x4 z4 = {0, 0, 0, 0};
  i32x8 z8 = {0, 0, 0, 0, 0, 0, 0, 0};
  __builtin_amdgcn_tensor_load_to_lds(g0, g1, z4, z4, z8, 0);
}
#endif

// ---------------------------------------------------------------- LayerNorm
__global__ __launch_bounds__(256) void ln_kernel(const float* __restrict__ x,
                                                 const float* __restrict__ g,
                                                 const float* __restrict__ be,
                                                 float* __restrict__ out) {
  int row = blockIdx.x;  // B*L rows
  int t = threadIdx.x;   // 256 == DIM
  float v = x[(long)row * DIM + t];
  __shared__ float red[256];
  red[t] = v;
  __syncthreads();
  for (int s = 128; s > 0; s >>= 1) {
    if (t < s) red[t] += red[t + s];
    __syncthreads();
  }
  float mu = red[0] * (1.0f / DIM);
  __syncthreads();
  float d = v - mu;
  red[t] = d * d;
  __syncthreads();
  for (int s = 128; s > 0; s >>= 1) {
    if (t < s) red[t] += red[t + s];
    __syncthreads();
  }
  float var = red[0] * (1.0f / DIM);
  float r = rsqrtf(var + 1e-5f);
  out[(long)row * DIM + t] = d * r * g[t] + be[t];
}

// ------------------------------------------- generic WMMA GEMM, 1 wave/tile
__global__ __launch_bounds__(32) void gemm16(const float* __restrict__ A,
                                             const float* __restrict__ W,
                                             float* __restrict__ C, int M,
                                             int N, int K,
                                             const float* __restrict__ rowScale) {
  int lane = threadIdx.x;
  int l16 = lane & 15;
  int half = lane >> 4;
  int n0 = blockIdx.x * 16;
  int m0 = blockIdx.y * 16;
  const float* Arow = A + (long)(m0 + l16) * K + 2 * half;
  const float* Wp = W + (long)(2 * half) * N + n0 + l16;
  v8f c = {};
  for (int kb = 0; kb < K; kb += 4) {
    v2f a;
    a.x = Arow[kb];
    a.y = Arow[kb + 1];
    v2f b;
    b.x = Wp[(long)kb * N];
    b.y = Wp[(long)kb * N + N];
    c = wmma4(a, b, c);
  }
#pragma unroll
  for (int r = 0; r < 8; ++r) {
    int row = m0 + half * 8 + r;
    float sc = rowScale ? rowScale[row] : 1.0f;
    C[(long)row * N + n0 + l16] = c[r] * sc;
  }
}

// ------------------------------------- bias = pair @ w_b  (the 536MB stream)
// TDM double-buffers 128-row chunks into LDS (pitch 260 => conflict-free
// b64 A-operand reads); 8 waves each run one 16-row WMMA tile per chunk.
#define BROWS 128
#define BPITCH 260  // 256 + 4 pad dwords (interval code 7, amount code 3)
#define BIAS_LDS_FLOATS (2 * BROWS * BPITCH + DIM * 16)

__global__ __launch_bounds__(256) void bias_kernel(const float* __restrict__ pair,
                                                   const float* __restrict__ wb,
                                                   float* __restrict__ bias) {
  extern __shared__ float blds[];
  float* buf0 = blds;
  float* buf1 = buf0 + BROWS * BPITCH;
  float* wbs = buf1 + BROWS * BPITCH;

  int tid = threadIdx.x;  // 256 = 8 waves
  for (int n = 0; n < 16; ++n)
    wbs[tid * 16 + n] = (n < NHEADS) ? wb[tid * NHEADS + n] : 0.0f;

  int lane = tid & 31, wave = tid >> 5;
  int l16 = lane & 15, half = lane >> 4;
  long chunk0 = (long)blockIdx.x * 8;  // 512 blocks * 8 chunks = 4096 chunks

#if USE_TDM
  if (wave == 0)
    tdm_load_2d(lds_of(buf0), pair + chunk0 * BROWS * DIM, DIM, BROWS, DIM,
                BROWS, DIM, 1, 7u, 3u);
#endif
  for (int c = 0; c < 8; ++c) {
    float* cur = (c & 1) ? buf1 : buf0;
#if USE_TDM
    float* nxt = (c & 1) ? buf0 : buf1;
    if (wave == 0) {
      if (c + 1 < 8) {
        tdm_load_2d(lds_of(nxt), pair + (chunk0 + c + 1) * BROWS * DIM, DIM,
                    BROWS, DIM, BROWS, DIM, 1, 7u, 3u);
        __builtin_amdgcn_s_wait_tensorcnt(1);  // oldest (chunk c) done
      } else {
        __builtin_amdgcn_s_wait_tensorcnt(0);  // last chunk done
      }
    }
    __syncthreads();
#else
    // Fallback: cooperative staging (coalesced b32 loads).
    const float* src = pair + (chunk0 + c) * BROWS * DIM;
    for (int idx = tid; idx < BROWS * DIM; idx += 256) {
      int rr = idx >> 8, cc = idx & 255;
      cur[rr * BPITCH + cc] = src[(long)rr * DIM + cc];
    }
    __syncthreads();
#endif
    const float* Arow = cur + (wave * 16 + l16) * BPITCH + 2 * half;
    v8f acc = {};
    for (int kb = 0; kb < DIM; kb += 4) {
      v2f a = *(const v2f*)&Arow[kb];
      int krow = kb + 2 * half;
      v2f b;
      b.x = wbs[krow * 16 + l16];
      b.y = wbs[krow * 16 + 16 + l16];
      acc = wmma4(a, b, acc);
    }
    if (l16 < NHEADS) {
      int h = l16;
      long m0 = (chunk0 + c) * BROWS + wave * 16;
#pragma unroll
      for (int r = 0; r < 8; ++r) {
        long row = m0 + half * 8 + r;  // b*L*L + i*L + j
        long b_ = row >> 18;
        long rem = row & (long)(LSEQ * LSEQ - 1);
        bias[(b_ * NHEADS + h) * (long)(LSEQ * LSEQ) + rem] = acc[r];
      }
    }
    __syncthreads();
  }
}

// ---------------------------------------------------------- fused attention
#define KP 34   // K pitch: TDM interval code 4 (32dw) + amount code 1 (2dw)
#define VP 40   // V pitch: TDM interval code 4 (32dw) + amount code 7 (8dw)
#define SP 516  // S pitch
#define ATTN_LDS_FLOATS (LSEQ * KP + LSEQ * VP + 16 * KP + 16 * SP + 8 * 256)

__global__ __launch_bounds__(256) void attn_kernel(
    const float* __restrict__ q, const float* __restrict__ kvbuf,
    const float* __restrict__ bias, const float* __restrict__ mask,
    float* __restrict__ ctx) {
  extern __shared__ float lds[];
  float* Kt = lds;             // LSEQ*KP
  float* Vt = Kt + LSEQ * KP;  // LSEQ*VP
  float* Qs = Vt + LSEQ * VP;  // 16*KP
  float* S = Qs + 16 * KP;     // 16*SP
  float* Pacc = S + 16 * SP;   // 8 waves * 16x16 partials

  int itile = blockIdx.x, h = blockIdx.y, b = blockIdx.z;
  int i0 = itile * 16;
  int tid = threadIdx.x;  // 256 = 8 waves
  int lane = tid & 31, wave = tid >> 5;
  int l16 = lane & 15, half = lane >> 4;

  const float* kbase = kvbuf + (long)b * LSEQ * 2 * DIM + h * HD;
  const float* vbase = kbase + DIM;
#if USE_TDM
  if (wave == 0) {
    tdm_load_2d(lds_of(Kt), kbase, HD, LSEQ, 2 * DIM, LSEQ, 2 * DIM, 1, 4u, 1u);
    tdm_load_2d(lds_of(Vt), vbase, HD, LSEQ, 2 * DIM, LSEQ, 2 * DIM, 1, 4u, 7u);
  }
#else
  for (int idx = tid; idx < LSEQ * HD; idx += 256) {
    int row = idx >> 5, col = idx & 31;
    Kt[row * KP + col] = kbase[(long)row * 2 * DIM + col];
    Vt[row * VP + col] = vbase[(long)row * 2 * DIM + col];
  }
#endif
  for (int idx = tid; idx < 16 * HD; idx += 256) {
    int row = idx >> 5, col = idx & 31;
    Qs[row * KP + col] = q[((long)b * LSEQ + i0 + row) * DIM + h * HD + col];
  }
#if USE_TDM
  if (wave == 0) __builtin_amdgcn_s_wait_tensorcnt(0);
#endif
  __syncthreads();

  // S = Q K^T * scale + bias, then mask.
  const float scl = 0.17677669529663687f;  // 1/sqrt(32)
  const float* biasBase = bias + ((long)(b * NHEADS + h) * LSEQ + i0) * LSEQ;
  for (int jt = wave; jt < LSEQ / 16; jt += 8) {
    int j0 = jt * 16;
    v8f c = {};
    for (int kb = 0; kb < HD; kb += 4) {
      v2f a = *(const v2f*)&Qs[l16 * KP + kb + 2 * half];
      v2f bb = *(const v2f*)&Kt[(j0 + l16) * KP + kb + 2 * half];
      c = wmma4(a, bb, c);
    }
    float m = mask[b * LSEQ + j0 + l16];
#pragma unroll
    for (int r = 0; r < 8; ++r) {
      int mrow = half * 8 + r;
      float val = c[r] * scl + biasBase[(long)mrow * LSEQ + j0 + l16];
      S[mrow * SP + j0 + l16] = (m != 0.0f) ? val : -1e9f;
    }
  }
  __syncthreads();

  // Row softmax: each wave owns 2 of the 16 rows.
  for (int rr = 0; rr < 2; ++rr) {
    int row = wave * 2 + rr;
    float vals[16];
    float mx = -3.0e38f;
#pragma unroll
    for (int t = 0; t < 16; ++t) {
      vals[t] = S[row * SP + lane + 32 * t];
      mx = fmaxf(mx, vals[t]);
    }
    for (int off = 16; off; off >>= 1) mx = fmaxf(mx, __shfl_xor(mx, off, 32));
    float sum = 0.f;
#pragma unroll
    for (int t = 0; t < 16; ++t) {
      vals[t] = __expf(vals[t] - mx);
      sum += vals[t];
    }
    for (int off = 16; off; off >>= 1) sum += __shfl_xor(sum, off, 32);
    float inv = 1.0f / sum;
#pragma unroll
    for (int t = 0; t < 16; ++t) S[row * SP + lane + 32 * t] = vals[t] * inv;
  }
  __syncthreads();

  // out = P @ V : waves split by (n-tile, K-chunk of 128).
  {
    int nt = wave & 1, kc0 = (wave >> 1) * 128;
    v8f c = {};
    for (int kb = kc0; kb < kc0 + 128; kb += 4) {
      v2f a = *(const v2f*)&S[l16 * SP + kb + 2 * half];
      int krow = kb + 2 * half;
      v2f bb;
      bb.x = Vt[krow * VP + nt * 16 + l16];
      bb.y = Vt[(krow + 1) * VP + nt * 16 + l16];
      c = wmma4(a, bb, c);
    }
#pragma unroll
    for (int r = 0; r < 8; ++r)
      Pacc[wave * 256 + (half * 8 + r) * 16 + l16] = c[r];
  }
  __syncthreads();
  for (int idx = tid; idx < 512; idx += 256) {
    int row = idx >> 5, col = idx & 31;
    int nt = col >> 4, cl = col & 15;
    float s = 0.f;
#pragma unroll
    for (int kc = 0; kc < 4; ++kc)
      s += Pacc[(kc * 2 + nt) * 256 + row * 16 + cl];
    ctx[((long)b * LSEQ + i0 + row) * DIM + h * HD + col] = s;
  }
}

// --------------------------------------------------------------------------
extern "C" void kernel_launch(void* const* d_in, const int* in_sizes, int n_in,
                              void* d_out, int out_size, void* d_ws,
                              size_t ws_size, hipStream_t stream) {
  (void)in_sizes; (void)n_in; (void)out_size; (void)ws_size;
  const float* single = (const float*)d_in[0];
  const float* pair   = (const float*)d_in[1];
  const float* mask   = (const float*)d_in[2];
  const float* w_q    = (const float*)d_in[3];
  const float* w_kv   = (const float*)d_in[4];
  const float* w_b    = (const float*)d_in[5];
  const float* w_out  = (const float*)d_in[6];
  const float* gamma  = (const float*)d_in[7];
  const float* beta   = (const float*)d_in[8];
  float* out = (float*)d_out;

  float* x   = (float*)d_ws;             // [B*L, D]
  float* qb  = x + BATCH * LSEQ * DIM;   // [B*L, D]
  float* kvb = qb + BATCH * LSEQ * DIM;  // [B*L, 2D]
  float* ctx = kvb + BATCH * LSEQ * 2 * DIM;
  float* biasbuf = ctx + BATCH * LSEQ * DIM;  // [B,H,L,L] 16 MB

  // 1) LayerNorm
  ln_kernel<<<BATCH * LSEQ, 256, 0, stream>>>(single, gamma, beta, x);
  // 2) projections
  gemm16<<<dim3(DIM / 16, BATCH * LSEQ / 16), 32, 0, stream>>>(
      x, w_q, qb, BATCH * LSEQ, DIM, DIM, nullptr);
  gemm16<<<dim3(2 * DIM / 16, BATCH * LSEQ / 16), 32, 0, stream>>>(
      x, w_kv, kvb, BATCH * LSEQ, 2 * DIM, DIM, nullptr);
  // 3) pair bias: TDM double-buffered 536MB stream through the matrix pipe
  const int bias_lds = BIAS_LDS_FLOATS * (int)sizeof(float);  // ~276 KB
  (void)hipFuncSetAttribute((const void*)bias_kernel,
                            hipFuncAttributeMaxDynamicSharedMemorySize,
                            bias_lds);
  bias_kernel<<<(BATCH * LSEQ * LSEQ / BROWS) / 8, 256, bias_lds, stream>>>(
      pair, w_b, biasbuf);
  // 4) fused attention (TDM-staged K/V)
  const int attn_lds = ATTN_LDS_FLOATS * (int)sizeof(float);  // ~195 KB
  (void)hipFuncSetAttribute((const void*)attn_kernel,
                            hipFuncAttributeMaxDynamicSharedMemorySize,
                            attn_lds);
  attn_kernel<<<dim3(LSEQ / 16, NHEADS, BATCH), 256, attn_lds, stream>>>(
      qb, kvb, biasbuf, mask, ctx);
  // 5) output projection with fused mask scale
  gemm16<<<dim3(DIM / 16, BATCH * LSEQ / 16), 32, 0, stream>>>(
      ctx, w_out, out, BATCH * LSEQ, DIM, DIM, mask);
}